// EmissionModel_77567109366255
// MI455X (gfx1250) — compile-verified
//
#include <hip/hip_runtime.h>
#include <hip/hip_bf16.h>
#include <math.h>

// Problem constants
#define CSZ 8
#define NVR 4
#define EDM 32
#define NCH 3
#define NB  128
#define NK  4096   // 8^4

typedef __attribute__((ext_vector_type(2))) float v2f;
typedef __attribute__((ext_vector_type(8))) float v8f;

// D = A(16x4 f32) * B(4x16 f32) + C(16x16 f32)
__device__ __forceinline__ v8f wmma_f32_k4(v2f a, v2f b, v8f c) {
  return __builtin_amdgcn_wmma_f32_16x16x4_f32(
      /*neg_a=*/false, a, /*neg_b=*/false, b,
      /*c_mod=*/(short)0, c, /*reuse_a=*/false, /*reuse_b=*/false);
}

// ---------------------------------------------------------------------------
// Stage 1: expand codebook -> Z (4096 x 128)
// ---------------------------------------------------------------------------
__global__ void build_z_kernel(const float* __restrict__ emb, float* __restrict__ Z) {
  int idx = blockIdx.x * blockDim.x + threadIdx.x;
  if (idx >= NK * (NVR * EDM)) return;
  int k = idx >> 7;
  int j = idx & 127;
  int v = j >> 5;
  int e = j & 31;
  int digit = (k >> (3 * (NVR - 1 - v))) & (CSZ - 1);
  Z[idx] = emb[(v * CSZ + digit) * EDM + e];
}

// ---------------------------------------------------------------------------
// Dense matrix of a 2x2 stride-1 VALID transposed conv (scatter semantics):
// Mt[(o,y,x)][(i,yi,xi)] = w[i][o][y-yi][x-xi]  if y-yi,x-xi in {0,1} else 0
// Rows = CO*(HI+1)*(WI+1), Cols = CI*HI*WI, row-major.
// ---------------------------------------------------------------------------
template <int CI, int CO, int HI, int WI>
__global__ void build_convmat_kernel(const float* __restrict__ w, float* __restrict__ Mt) {
  constexpr int HO = HI + 1, WO = WI + 1;
  constexpr int COLS = CI * HI * WI;
  constexpr int ROWS = CO * HO * WO;
  int idx = blockIdx.x * blockDim.x + threadIdx.x;
  if (idx >= ROWS * COLS) return;
  int col = idx % COLS;
  int row = idx / COLS;
  int xi = col % WI; int t = col / WI;
  int yi = t % HI;   int i = t / HI;
  int x = row % WO;  int t2 = row / WO;
  int y = t2 % HO;   int o = t2 / HO;
  int a = y - yi, b = x - xi;
  float val = 0.0f;
  if (a >= 0 && a < 2 && b >= 0 && b < 2)
    val = w[((i * CO + o) * 2 + a) * 2 + b];
  Mt[idx] = val;
}

// ---------------------------------------------------------------------------
// Composed bias: out[r] = badd[r / sp_out] + sum_l M[r][l] * vin[l / sp_in]
// ---------------------------------------------------------------------------
__global__ void bias_compose_kernel(const float* __restrict__ M, int L,
                                    const float* __restrict__ vin, int sp_in,
                                    const float* __restrict__ badd, int sp_out,
                                    float* __restrict__ out, int R) {
  int r = blockIdx.x * blockDim.x + threadIdx.x;
  if (r >= R) return;
  const float* row = M + (long)r * L;
  float s = badd[r / sp_out];
  for (int l = 0; l < L; ++l) s += row[l] * vin[l / sp_in];
  out[r] = s;
}

// ---------------------------------------------------------------------------
// WMMA f32 GEMM, 16x64 tile per wave (4 accumulators share one A fragment).
//   B_TRANSPOSED=1:  C[MxN] = A[MxKd] @ Bt[NxKd]^T       (Bt row-major)
//   B_TRANSPOSED=0:  C[MxN] = A[MxKd] @ B[KdxN]          (B  row-major)
// Epilogue: + bias[n / bias_div] (bias may be null), optional ELU (act==1).
// Kd % 4 == 0. M, N arbitrary: out-of-range lanes are pointer-clamped (their
// garbage lands only in never-stored rows/columns), stores fully guarded.
// ---------------------------------------------------------------------------
template <bool B_TRANSPOSED>
__global__ void gemm64_wmma_f32_kernel(const float* __restrict__ A, int lda,
                                       const float* __restrict__ B, int ldb,
                                       const float* __restrict__ bias, int bias_div,
                                       float* __restrict__ C, int ldc,
                                       int M, int N, int Kd, int act) {
  int tilesN = (N + 63) >> 6;
  int tilesM = (M + 15) >> 4;
  int wave = (blockIdx.x * blockDim.x + threadIdx.x) >> 5;
  int lane = threadIdx.x & 31;
  int tm = wave / tilesN;
  int tn = wave - tm * tilesN;
  if (tm >= tilesM) return;                     // uniform across wave

  int m0 = tm << 4, n0 = tn << 6;
  int r  = lane & 15;
  int hi = lane >> 4;                           // 0: K=kk..kk+1, 1: K=kk+2..kk+3

  int arowi = m0 + r; if (arowi > M - 1) arowi = M - 1;          // clamp
  const float* arow = A + (long)arowi * lda;

  int   ncol[4];
  const float* bptr[4];
#pragma unroll
  for (int j = 0; j < 4; ++j) {
    ncol[j] = n0 + 16 * j + r;
    int nc = ncol[j] < N ? ncol[j] : (N - 1);                    // clamp
    bptr[j] = B_TRANSPOSED ? (B + (long)nc * ldb) : (B + nc);
  }

  v8f acc[4] = {v8f{}, v8f{}, v8f{}, v8f{}};
  for (int kk = 0; kk < Kd; kk += 4) {
    int k0 = kk + 2 * hi;
    v2f a;
    a.x = arow[k0];
    a.y = arow[k0 + 1];
    v2f b[4];
#pragma unroll
    for (int j = 0; j < 4; ++j) {
      if (B_TRANSPOSED) {
        b[j].x = bptr[j][k0];
        b[j].y = bptr[j][k0 + 1];
      } else {
        b[j].x = bptr[j][(long)k0 * ldb];
        b[j].y = bptr[j][(long)(k0 + 1) * ldb];
      }
    }
#pragma unroll
    for (int j = 0; j < 4; ++j) acc[j] = wmma_f32_k4(a, b[j], acc[j]);
  }

#pragma unroll
  for (int j = 0; j < 4; ++j) {
    if (ncol[j] < N) {
      float bv = bias ? bias[ncol[j] / bias_div] : 0.0f;
#pragma unroll
      for (int v = 0; v < 8; ++v) {
        int mrow = m0 + v + 8 * hi;             // VGPR v -> M = v (+8 for hi lanes)
        if (mrow < M) {
          float s = acc[j][v] + bv;
          if (act == 1) s = (s > 0.0f) ? s : expm1f(s);
          C[(long)mrow * ldc + ncol[j]] = s;
        }
      }
    }
  }
}

// ---------------------------------------------------------------------------
// Row norms over 49-element spatial slices: dst[r][c] = sum_p src[r][c][p]^2
// ---------------------------------------------------------------------------
__global__ void rownorm_kernel(const float* __restrict__ src, float* __restrict__ dst,
                               int nrows) {
  int idx = blockIdx.x * blockDim.x + threadIdx.x;
  if (idx >= nrows * NCH) return;
  int c = idx % NCH;
  int r = idx / NCH;
  const float* p = src + (long)r * (NCH * 49) + c * 49;
  float s = 0.0f;
  for (int i = 0; i < 49; ++i) s += p[i] * p[i];
  dst[idx] = s;
}

// ---------------------------------------------------------------------------
// logits[b,c,k] = <x_bc, m_kc> - 0.5*(||m_kc||^2 + ||x_bc||^2)
// Kd=49 padded to 52 with exec-masked zero loads (contraction-dim padding
// must be exact zeros).
// ---------------------------------------------------------------------------
__global__ void logits_wmma_kernel(const float* __restrict__ X,      // (128,3,49)
                                   const float* __restrict__ MEANS,  // (4096,3,49)
                                   const float* __restrict__ M2,     // (4096,3)
                                   const float* __restrict__ X2,     // (128,3)
                                   float* __restrict__ out) {        // (128,3,4096)
  int c = blockIdx.z;
  const int tilesN = NK >> 4;                  // 256
  int wave = (blockIdx.x * blockDim.x + threadIdx.x) >> 5;
  int lane = threadIdx.x & 31;
  int tm = wave / tilesN;                      // 0..7
  int tn = wave - tm * tilesN;
  if (tm >= (NB >> 4)) return;

  int m0 = tm << 4, n0 = tn << 4;
  int r  = lane & 15;
  int hi = lane >> 4;

  const float* arow = X     + (long)(m0 + r) * (NCH * 49) + c * 49;
  const float* brow = MEANS + (long)(n0 + r) * (NCH * 49) + c * 49;

  v8f acc = {};
  for (int kk = 0; kk < 52; kk += 4) {
    int k0 = kk + 2 * hi;
    v2f a, b;
    a.x = (k0     < 49) ? arow[k0]     : 0.0f;
    a.y = (k0 + 1 < 49) ? arow[k0 + 1] : 0.0f;
    b.x = (k0     < 49) ? brow[k0]     : 0.0f;
    b.y = (k0 + 1 < 49) ? brow[k0 + 1] : 0.0f;
    acc = wmma_f32_k4(a, b, acc);
  }

  int ncol = n0 + r;
  float mh = M2[ncol * NCH + c];
#pragma unroll
  for (int v = 0; v < 8; ++v) {
    int brow_i = m0 + v + 8 * hi;
    out[(long)brow_i * (NCH * NK) + (long)c * NK + ncol] =
        acc[v] - 0.5f * (mh + X2[brow_i * NCH + c]);
  }
}

// ---------------------------------------------------------------------------
extern "C" void kernel_launch(void* const* d_in, const int* in_sizes, int n_in,
                              void* d_out, int out_size, void* d_ws, size_t ws_size,
                              hipStream_t stream) {
  (void)in_sizes; (void)n_in; (void)out_size; (void)ws_size;

  const float* x     = (const float*)d_in[0];   // (128,3,7,7)
  const float* emb   = (const float*)d_in[1];   // (4,8,32)
  const float* w_fc1 = (const float*)d_in[2];   // (1024,128)
  const float* b_fc1 = (const float*)d_in[3];
  const float* w_ct1 = (const float*)d_in[4];   // (64,32,2,2)
  const float* b_ct1 = (const float*)d_in[5];
  const float* w_ct2 = (const float*)d_in[6];   // (32,16,2,2)
  const float* b_ct2 = (const float*)d_in[7];
  const float* w_ct3 = (const float*)d_in[8];   // (16,16,2,2)
  const float* b_ct3 = (const float*)d_in[9];
  const float* w_fc2 = (const float*)d_in[10];  // (147,784)
  const float* b_fc2 = (const float*)d_in[11];
  float* out = (float*)d_out;

  // Workspace layout (floats), all regions distinct (~34.8 MB total)
  float* ws    = (float*)d_ws;
  float* Z     = ws;                      // 4096*128          =   524288
  float* M1    = Z     +  524288;         // 800*1024          =   819200
  float* W_A   = M1    +  819200;         // 800*128           =   102400
  float* BIASA = W_A   +  102400;         // 800
  float* M2m   = BIASA +     800;         // 576*800           =   460800
  float* M3m   = M2m   +  460800;         // 784*576           =   451584
  float* W_C   = M3m   +  451584;         // 147*576           =    84672
  float* BIASC = W_C   +   84672;         // 147
  float* A1    = BIASC +     147;         // 4096*800          =  3276800
  float* A2    = A1    + 3276800;         // 4096*576          =  2359296
  float* MEANS = A2    + 2359296;         // 4096*147          =   602112
  float* M2n   = MEANS +  602112;         // 4096*3            =    12288
  float* X2n   = M2n   +   12288;         // 128*3

  dim3 blk(256);

  // --- codebook gather ------------------------------------------------------
  build_z_kernel<<<(NK * 128) / 256, blk, 0, stream>>>(emb, Z);

  // --- build dense conv matrices -------------------------------------------
  build_convmat_kernel<64, 32, 4, 4><<<(800 * 1024) / 256, blk, 0, stream>>>(w_ct1, M1);
  build_convmat_kernel<32, 16, 5, 5><<<(576 * 800) / 256, blk, 0, stream>>>(w_ct2, M2m);
  build_convmat_kernel<16, 16, 6, 6><<<(784 * 576 + 255) / 256, blk, 0, stream>>>(w_ct3, M3m);

  // --- compose ct1*fc1:  W_A(800x128) = M1(800x1024) @ w_fc1(1024x128) ------
  {
    int waves = 50 * 2;  // tilesM(800)=50, tilesN64(128)=2
    gemm64_wmma_f32_kernel<false><<<(waves * 32 + 255) / 256, blk, 0, stream>>>(
        M1, 1024, w_fc1, 128, nullptr, 1, W_A, 128, 800, 128, 1024, 0);
  }
  // biasA[r] = b_ct1[r/25] + M1 @ b_fc1
  bias_compose_kernel<<<(800 + 255) / 256, blk, 0, stream>>>(
      M1, 1024, b_fc1, 1, b_ct1, 25, BIASA, 800);

  // --- compose fc2*ct3:  W_C(147x576) = w_fc2(147x784) @ M3(784x576) --------
  {
    int waves = 10 * 9;  // tilesM(147)=10, tilesN64(576)=9
    gemm64_wmma_f32_kernel<false><<<(waves * 32 + 255) / 256, blk, 0, stream>>>(
        w_fc2, 784, M3m, 576, nullptr, 1, W_C, 576, 147, 576, 784, 0);
  }
  // biasC[n] = b_fc2[n] + w_fc2 @ b_ct3_expanded
  bias_compose_kernel<<<(147 + 255) / 256, blk, 0, stream>>>(
      w_fc2, 784, b_ct3, 49, b_fc2, 1, BIASC, 147);

  // --- A1 = ELU( Z @ W_A^T + biasA )   (4096x800, Kd=128) -------------------
  {
    int waves = 256 * 13;  // tilesM=256, tilesN64(800)=13
    gemm64_wmma_f32_kernel<true><<<(waves * 32 + 255) / 256, blk, 0, stream>>>(
        Z, 128, W_A, 128, BIASA, 1, A1, 800, NK, 800, 128, 1);
  }
  // --- A2 = ELU( A1 @ M2^T + b_ct2 )   (4096x576, Kd=800) -------------------
  {
    int waves = 256 * 9;   // tilesN64(576)=9
    gemm64_wmma_f32_kernel<true><<<(waves * 32 + 255) / 256, blk, 0, stream>>>(
        A1, 800, M2m, 800, b_ct2, 36, A2, 576, NK, 576, 800, 1);
  }
  // --- means = A2 @ W_C^T + biasC      (4096x147, Kd=576) -------------------
  {
    int waves = 256 * 3;   // tilesN64(147)=3
    gemm64_wmma_f32_kernel<true><<<(waves * 32 + 255) / 256, blk, 0, stream>>>(
        A2, 576, W_C, 576, BIASC, 1, MEANS, 147, NK, 147, 576, 0);
  }

  // --- squared norms --------------------------------------------------------
  rownorm_kernel<<<(NK * NCH + 255) / 256, blk, 0, stream>>>(MEANS, M2n, NK);
  rownorm_kernel<<<(NB * NCH + 255) / 256, blk, 0, stream>>>(x, X2n, NB);

  // --- logits: per-channel 128x4096x49 WMMA GEMM + norm combine -------------
  logits_wmma_kernel<<<dim3(256, 1, 3), blk, 0, stream>>>(x, MEANS, M2n, X2n, out);
}